// JVPCirculant_23330262352114
// MI455X (gfx1250) — compile-verified
//
#include <hip/hip_runtime.h>
#include <hip/hip_bf16.h>

typedef __attribute__((ext_vector_type(2))) float v2f;
typedef __attribute__((ext_vector_type(8))) float v8f;

#define NFFT  4096
#define LDIM  64
#define TWO_PI 6.28318530717958647692f

// ---------------- ws layout (floats) ----------------
#define WS_COS 0        // COS64[i*64+j] = cos(2*pi*i*j/64)
#define WS_SIN 4096     // SIN64
#define WS_TWC 8192     // cos(2*pi*i*j/4096)
#define WS_TWS 12288    // sin(2*pi*i*j/4096)
#define WS_HRE 16384    // H real,  [k1*64+k2]
#define WS_HIM 20480    // H imag
#define WS_GTR 24576    // kernel1 scratch
#define WS_GTI 28672

// ---------------- WMMA fragment helpers (f32 16x16x4, wave32) ----------------
__device__ __forceinline__ v2f fragA(const float* m, int r0, int k0, int lane) {
    int row = r0 + (lane & 15);
    int kk  = k0 + ((lane >> 4) << 1);
    v2f a;
    a.x = m[row * LDIM + kk];
    a.y = m[row * LDIM + kk + 1];
    return a;
}
__device__ __forceinline__ v2f fragB(const float* m, int k0, int c0, int lane) {
    int col = c0 + (lane & 15);
    int kk  = k0 + ((lane >> 4) << 1);
    v2f b;
    b.x = m[kk * LDIM + col];
    b.y = m[(kk + 1) * LDIM + col];
    return b;
}
__device__ __forceinline__ void storeC(float* m, v8f acc, int r0, int c0, int lane) {
    int col  = c0 + (lane & 15);
    int roff = r0 + ((lane >> 4) << 3);
#pragma unroll
    for (int v = 0; v < 8; ++v) m[(roff + v) * LDIM + col] = acc[v];
}

// Two 16x16 output tiles (same row block r0, columns c0 and c0+16) of
//   OUT = (NEG1? -A1:A1)*B1  [+ (NEG2? -A2:A2)*B2]
// A-fragments are shared between the two tiles; two independent acc chains.
template <bool DUAL, bool NEG1, bool NEG2>
__device__ __forceinline__ void mmtile2(v8f& acc0, v8f& acc1,
                                        const float* A1, const float* B1,
                                        const float* A2, const float* B2,
                                        int r0, int c0, int lane) {
    acc0 = (v8f){};
    acc1 = (v8f){};
    const int c1 = c0 + 16;
#pragma unroll
    for (int kt = 0; kt < 16; ++kt) {
        int k0 = kt * 4;
        v2f a = fragA(A1, r0, k0, lane);
        if (NEG1) a = -a;
        v2f b0 = fragB(B1, k0, c0, lane);
        v2f b1 = fragB(B1, k0, c1, lane);
        acc0 = __builtin_amdgcn_wmma_f32_16x16x4_f32(false, a, false, b0,
                                                     (short)0, acc0, false, false);
        acc1 = __builtin_amdgcn_wmma_f32_16x16x4_f32(false, a, false, b1,
                                                     (short)0, acc1, false, false);
        if (DUAL) {
            v2f a2 = fragA(A2, r0, k0, lane);
            if (NEG2) a2 = -a2;
            v2f b20 = fragB(B2, k0, c0, lane);
            v2f b21 = fragB(B2, k0, c1, lane);
            acc0 = __builtin_amdgcn_wmma_f32_16x16x4_f32(false, a2, false, b20,
                                                         (short)0, acc0, false, false);
            acc1 = __builtin_amdgcn_wmma_f32_16x16x4_f32(false, a2, false, b21,
                                                         (short)0, acc1, false, false);
        }
    }
}

// Async 128-bit DMA global -> LDS (gfx1250), tracked by ASYNCcnt.
// ldsoff = wave-relative LDS byte address (low 32 bits of generic LDS pointer).
__device__ __forceinline__ void async_g2l_b128(unsigned ldsoff, const void* gaddr) {
    asm volatile("global_load_async_to_lds_b128 %0, %1, off"
                 :
                 : "v"(ldsoff), "v"((unsigned long long)(size_t)gaddr)
                 : "memory");
}
__device__ __forceinline__ void wait_asynccnt0() {
    asm volatile("s_wait_asynccnt 0" ::: "memory");
}

// ---------------- kernel 0: trig tables ----------------
__global__ void __launch_bounds__(256) circ_tables(float* ws) {
    int e = blockIdx.x * 256 + threadIdx.x;   // 0..4095
    int i = e >> 6, j = e & 63;
    int p = i * j;
    float s64, c64, s4k, c4k;
    __sincosf((TWO_PI / 64.0f)   * (float)(p & 63),   &s64, &c64);
    __sincosf((TWO_PI / 4096.0f) * (float)(p & 4095), &s4k, &c4k);
    ws[WS_COS + e] = c64;
    ws[WS_SIN + e] = s64;
    ws[WS_TWC + e] = c4k;
    ws[WS_TWS + e] = s4k;
}

// ---------------- kernel 1a: stage-1 of FFT(first_col) ----------------
__global__ void __launch_bounds__(256) circ_hstage1(const float* __restrict__ fr,
                                                    float* __restrict__ ws) {
    int e = blockIdx.x * 256 + threadIdx.x;   // 0..4095
    int k1 = e >> 6, n1 = e & 63;
    float gr = 0.f, gi = 0.f;
    for (int n2 = 0; n2 < 64; ++n2) {
        int idx = n1 + (n2 << 6);
        float cv  = (idx == 0) ? fr[0] : fr[NFFT - idx];
        float wc  = ws[WS_COS + (k1 << 6) + n2];
        float wsn = ws[WS_SIN + (k1 << 6) + n2];
        gr += cv * wc;
        gi -= cv * wsn;
    }
    float tc = ws[WS_TWC + (k1 << 6) + n1];
    float ts = ws[WS_TWS + (k1 << 6) + n1];
    ws[WS_GTR + e] = gr * tc + gi * ts;   // *(c - i s)
    ws[WS_GTI + e] = gi * tc - gr * ts;
}

// ---------------- kernel 1b: stage-2 -> H[k1][k2] ----------------
__global__ void __launch_bounds__(256) circ_hstage2(float* __restrict__ ws) {
    int e = blockIdx.x * 256 + threadIdx.x;   // e = k1*64 + k2
    int k1 = e >> 6, k2 = e & 63;
    float hr = 0.f, hi = 0.f;
    for (int n1 = 0; n1 < 64; ++n1) {
        float gr  = ws[WS_GTR + (k1 << 6) + n1];
        float gi  = ws[WS_GTI + (k1 << 6) + n1];
        float wc  = ws[WS_COS + (k2 << 6) + n1];
        float wsn = ws[WS_SIN + (k2 << 6) + n1];
        hr += gr * wc + gi * wsn;
        hi += gi * wc - gr * wsn;
    }
    ws[WS_HRE + e] = hr;
    ws[WS_HIM + e] = hi;
}

// ---------------- main kernel: one batch row per workgroup ----------------
// LDS (floats): COS 0 | SIN 4096 | TWC 8192 | TWS 12288 | R 16384 |
//               Gr 20480 | Gi 24576 | Xr 28672 | Xi 32768  => 36864 f = 144KB
__global__ void __launch_bounds__(256) circ_main(const float* __restrict__ x,
                                                 const float* __restrict__ bias,
                                                 const float* __restrict__ ws,
                                                 float* __restrict__ y) {
    extern __shared__ float smem[];
    float* cosT = smem;
    float* sinT = smem + 4096;
    float* twc  = smem + 8192;
    float* tws  = smem + 12288;
    float* R    = smem + 16384;
    float* Gr   = smem + 20480;
    float* Gi   = smem + 24576;
    float* Xr   = smem + 28672;
    float* Xi   = smem + 32768;

    const int tid  = threadIdx.x;
    const int lane = tid & 31;
    const int wave = tid >> 5;
    const int b    = blockIdx.x;

    // ---- async DMA: 4 trig tables (16384 f, contiguous) + x row into LDS ----
    {
        const unsigned lds0 = (unsigned)(size_t)(void*)smem;      // LDS byte offset
        const char*    gt   = (const char*)ws;
        const char*    gx   = (const char*)(x + (size_t)b * NFFT);
#pragma unroll
        for (int i = 0; i < 16; ++i) {                            // 64KB tables
            unsigned off = (unsigned)(tid + i * 256) * 16u;
            async_g2l_b128(lds0 + off, gt + off);
        }
#pragma unroll
        for (int i = 0; i < 4; ++i) {                             // 16KB row
            unsigned off = (unsigned)(tid + i * 256) * 16u;
            async_g2l_b128(lds0 + 16384u * 4u + off, gx + off);
        }
        wait_asynccnt0();
    }
    __syncthreads();

    const int r0 = (wave >> 1) << 4;      // row block 0..3
    const int c0 = (wave & 1) << 5;       // column pair base: 0 or 32
    v8f a0, a1;

    // Stage 1: Gr = COS*R ; Gi = -SIN*R          (G = W64 * R, R real)
    mmtile2<false, false, false>(a0, a1, cosT, R, nullptr, nullptr, r0, c0, lane);
    storeC(Gr, a0, r0, c0, lane); storeC(Gr, a1, r0, c0 + 16, lane);
    mmtile2<false, true,  false>(a0, a1, sinT, R, nullptr, nullptr, r0, c0, lane);
    storeC(Gi, a0, r0, c0, lane); storeC(Gi, a1, r0, c0 + 16, lane);
    __syncthreads();

    // Forward twiddle: (Gr + iGi) *= (c - i s)
    for (int e = tid; e < 4096; e += 256) {
        float gr = Gr[e], gi = Gi[e], c = twc[e], s = tws[e];
        Gr[e] = gr * c + gi * s;
        Gi[e] = gi * c - gr * s;
    }
    __syncthreads();

    // Stage 2: Xr = Gr*COS + Gi*SIN ; Xi = Gi*COS - Gr*SIN     (X = Gt * W64)
    mmtile2<true, false, false>(a0, a1, Gr, cosT, Gi, sinT, r0, c0, lane);
    storeC(Xr, a0, r0, c0, lane); storeC(Xr, a1, r0, c0 + 16, lane);
    mmtile2<true, false, true >(a0, a1, Gi, cosT, Gr, sinT, r0, c0, lane);
    storeC(Xi, a0, r0, c0, lane); storeC(Xi, a1, r0, c0 + 16, lane);
    __syncthreads();

    // Pointwise multiply by H (L2-resident)
    {
        const float* Hre = ws + WS_HRE;
        const float* Him = ws + WS_HIM;
        for (int e = tid; e < 4096; e += 256) {
            float xr = Xr[e], xi = Xi[e];
            float hr = Hre[e], hi = Him[e];
            Xr[e] = xr * hr - xi * hi;
            Xi[e] = xr * hi + xi * hr;
        }
    }
    __syncthreads();

    // Stage 3: Gr = Xr*COS - Xi*SIN ; Gi = Xr*SIN + Xi*COS    (G2 = Y * conj(W64))
    mmtile2<true, false, true >(a0, a1, Xr, cosT, Xi, sinT, r0, c0, lane);
    storeC(Gr, a0, r0, c0, lane); storeC(Gr, a1, r0, c0 + 16, lane);
    mmtile2<true, false, false>(a0, a1, Xr, sinT, Xi, cosT, r0, c0, lane);
    storeC(Gi, a0, r0, c0, lane); storeC(Gi, a1, r0, c0 + 16, lane);
    __syncthreads();

    // Inverse twiddle: *= (c + i s)
    for (int e = tid; e < 4096; e += 256) {
        float gr = Gr[e], gi = Gi[e], c = twc[e], s = tws[e];
        Gr[e] = gr * c - gi * s;
        Gi[e] = gi * c + gr * s;
    }
    __syncthreads();

    // Stage 4: out = (COS*Gr - SIN*Gi)/N + bias, direct to global
    {
        const float inv = 1.0f / (float)NFFT;
        mmtile2<true, false, true>(a0, a1, cosT, Gr, sinT, Gi, r0, c0, lane);
        int roff = r0 + ((lane >> 4) << 3);
#pragma unroll
        for (int v = 0; v < 8; ++v) {
            int n0 = (roff + v) * LDIM + c0 + (lane & 15);        // n = n1 + 64*n2
            int n1i = n0 + 16;
            y[(size_t)b * NFFT + n0]  = a0[v] * inv + bias[n0];
            y[(size_t)b * NFFT + n1i] = a1[v] * inv + bias[n1i];
        }
    }
}

extern "C" void kernel_launch(void* const* d_in, const int* in_sizes, int n_in,
                              void* d_out, int out_size, void* d_ws, size_t ws_size,
                              hipStream_t stream) {
    const float* x    = (const float*)d_in[0];
    const float* fr   = (const float*)d_in[1];
    const float* bias = (const float*)d_in[2];
    float*       y    = (float*)d_out;
    float*       ws   = (float*)d_ws;

    int batch = out_size / NFFT;   // 8192

    circ_tables <<<16, 256, 0, stream>>>(ws);
    circ_hstage1<<<16, 256, 0, stream>>>(fr, ws);
    circ_hstage2<<<16, 256, 0, stream>>>(ws);
    circ_main   <<<batch, 256, 36864 * sizeof(float), stream>>>(x, bias, ws, y);

    (void)in_sizes; (void)n_in; (void)ws_size;
}